// gat_full_10_51994874085800
// MI455X (gfx1250) — compile-verified
//
#include <hip/hip_runtime.h>
#include <hip/hip_bf16.h>

typedef __attribute__((ext_vector_type(2))) float v2f;
typedef __attribute__((ext_vector_type(8))) float v8f;

#define NNODES 50000
#define CFEAT  64
#define E0     1000000
#define ETOT   (E0 + NNODES)
#define NEG_SLOPE 0.2f

// ---------------- edge list build: int -> int32 src/dst with self loops ----
__global__ __launch_bounds__(256) void build_edges(const int* __restrict__ ei,
                                                   int* __restrict__ src,
                                                   int* __restrict__ dst) {
    int e = blockIdx.x * 256 + threadIdx.x;
    if (e >= ETOT) return;
    if (e < E0) { src[e] = ei[e]; dst[e] = ei[E0 + e]; }
    else        { int n = e - E0; src[e] = n; dst[e] = n; }
}

// ---------------- GEMM: H = X @ W via fp32 WMMA 16x16x4 --------------------
// One wave computes a 16-row x 64-col strip of H. 3125 waves cover 50000 rows.
__global__ __launch_bounds__(256) void gemm_h(const float* __restrict__ X,
                                              const float* __restrict__ W,
                                              float* __restrict__ H) {
    int wave = blockIdx.x * 8 + (threadIdx.x >> 5);
    int lane = threadIdx.x & 31;
    int row0 = wave * 16;
    if (row0 >= NNODES) return;          // uniform per wave; EXEC stays all-1s
    int lhalf = lane >> 4;               // 0: K pair {0,1}; 1: K pair {2,3}
    int lmod  = lane & 15;
    int arow  = row0 + lmod;             // A-matrix row for this lane

    v8f acc0 = {}, acc1 = {}, acc2 = {}, acc3 = {};
    for (int k0 = 0; k0 < CFEAT; k0 += 4) {
        int ka = k0 + 2 * lhalf;
        // A fragment: 2 consecutive K values of row `arow`
        v2f a = *(const v2f*)(X + (size_t)arow * CFEAT + ka);
        // B fragments: rows ka, ka+1 of W at 4 column tiles
        v2f b0, b1, b2, b3;
        b0.x = W[ka * CFEAT +  0 + lmod]; b0.y = W[(ka + 1) * CFEAT +  0 + lmod];
        b1.x = W[ka * CFEAT + 16 + lmod]; b1.y = W[(ka + 1) * CFEAT + 16 + lmod];
        b2.x = W[ka * CFEAT + 32 + lmod]; b2.y = W[(ka + 1) * CFEAT + 32 + lmod];
        b3.x = W[ka * CFEAT + 48 + lmod]; b3.y = W[(ka + 1) * CFEAT + 48 + lmod];
        acc0 = __builtin_amdgcn_wmma_f32_16x16x4_f32(false, a, false, b0, (short)0, acc0, false, false);
        acc1 = __builtin_amdgcn_wmma_f32_16x16x4_f32(false, a, false, b1, (short)0, acc1, false, false);
        acc2 = __builtin_amdgcn_wmma_f32_16x16x4_f32(false, a, false, b2, (short)0, acc2, false, false);
        acc3 = __builtin_amdgcn_wmma_f32_16x16x4_f32(false, a, false, b3, (short)0, acc3, false, false);
    }
    // D layout: VGPR r -> row row0 + r + 8*lhalf, col = tile*16 + lmod
    #pragma unroll
    for (int r = 0; r < 8; ++r) {
        size_t row = (size_t)(row0 + r + 8 * lhalf);
        float* out = H + row * CFEAT + lmod;
        out[ 0] = acc0[r];
        out[16] = acc1[r];
        out[32] = acc2[r];
        out[48] = acc3[r];
    }
}

// ------------- per-node: alpha_src/alpha_dst dots; init m, s, accum --------
__global__ __launch_bounds__(256) void node_prep(const float* __restrict__ H,
                                                 const float* __restrict__ a_src,
                                                 const float* __restrict__ a_dst,
                                                 float* __restrict__ as,
                                                 float* __restrict__ ad,
                                                 unsigned* __restrict__ mU,
                                                 float* __restrict__ sden,
                                                 float* __restrict__ accum) {
    int n = blockIdx.x * 256 + threadIdx.x;
    if (n >= NNODES) return;
    const float4* h4 = (const float4*)(H + (size_t)n * CFEAT);
    const float4* s4 = (const float4*)a_src;
    const float4* d4 = (const float4*)a_dst;
    float4* acc4 = (float4*)(accum + (size_t)n * CFEAT);
    float sa = 0.f, sd = 0.f;
    #pragma unroll
    for (int c = 0; c < CFEAT / 4; ++c) {
        float4 h = h4[c], A = s4[c], D = d4[c];
        sa += h.x * A.x + h.y * A.y + h.z * A.z + h.w * A.w;
        sd += h.x * D.x + h.y * D.y + h.z * D.z + h.w * D.w;
        acc4[c] = make_float4(0.f, 0.f, 0.f, 0.f);
    }
    as[n] = sa;
    ad[n] = sd;
    mU[n] = 0x007FFFFFu;   // monotonic-uint encoding of -inf
    sden[n] = 0.f;
}

// monotonic f32<->u32 mapping for atomic max on floats
__device__ __forceinline__ unsigned f2mono(float f) {
    unsigned u = __float_as_uint(f);
    return (u & 0x80000000u) ? ~u : (u | 0x80000000u);
}
__device__ __forceinline__ float mono2f(unsigned u) {
    return (u & 0x80000000u) ? __uint_as_float(u & 0x7FFFFFFFu)
                             : __uint_as_float(~u);
}

// ---------------- edge pass 1: segment max of leaky-relu logits ------------
__global__ __launch_bounds__(256) void edge_max(const int* __restrict__ src,
                                                const int* __restrict__ dst,
                                                const float* __restrict__ as,
                                                const float* __restrict__ ad,
                                                unsigned* __restrict__ mU) {
    int e = blockIdx.x * 256 + threadIdx.x;
    if (e >= ETOT) return;
    int dv = dst[e];
    float a = as[src[e]] + ad[dv];
    a = a > 0.f ? a : NEG_SLOPE * a;
    atomicMax(&mU[dv], f2mono(a));
}

// ------ edge pass 2 (fused): e = exp(a-m); s[dst]+=e; accum[dst]+=h[src]*e -
// One wave (32 lanes) per edge; each lane handles 2 features via float2.
__global__ __launch_bounds__(256) void edge_scatter(const int* __restrict__ src,
                                                    const int* __restrict__ dst,
                                                    const float* __restrict__ as,
                                                    const float* __restrict__ ad,
                                                    const unsigned* __restrict__ mU,
                                                    const float* __restrict__ H,
                                                    float* __restrict__ sden,
                                                    float* __restrict__ accum) {
    long long gid = (long long)blockIdx.x * 256 + threadIdx.x;
    int e = (int)(gid >> 5);
    int lane = threadIdx.x & 31;
    if (e >= ETOT) return;
    int sv = src[e], dv = dst[e];
    // same-address loads across the wave coalesce to a broadcast
    float a = as[sv] + ad[dv];
    a = a > 0.f ? a : NEG_SLOPE * a;
    float m = mono2f(mU[dv]);
    float ev = __expf(a - m);
    if (lane == 0) atomicAdd(&sden[dv], ev);
    float2 hv = *(const float2*)(H + (size_t)sv * CFEAT + 2 * lane);
    float* acc = accum + (size_t)dv * CFEAT + 2 * lane;
    atomicAdd(acc + 0, hv.x * ev);
    atomicAdd(acc + 1, hv.y * ev);
}

// ---------------- epilogue: x' = elu(accum / s + b) ------------------------
__global__ __launch_bounds__(256) void finalize(const float* __restrict__ accum,
                                                const float* __restrict__ sden,
                                                const float* __restrict__ bias,
                                                float* __restrict__ xout) {
    int gid = blockIdx.x * 256 + threadIdx.x;
    if (gid >= NNODES * CFEAT) return;
    int n = gid >> 6, c = gid & 63;
    float v = accum[gid] / sden[n] + bias[c];
    xout[gid] = v > 0.f ? v : (__expf(v) - 1.f);
}

extern "C" void kernel_launch(void* const* d_in, const int* in_sizes, int n_in,
                              void* d_out, int out_size, void* d_ws, size_t ws_size,
                              hipStream_t stream) {
    const float* x_in   = (const float*)d_in[0];
    const int*   ei     = (const int*)d_in[1];
    const float* W[2]   = { (const float*)d_in[2], (const float*)d_in[6] };
    const float* Asrc[2]= { (const float*)d_in[3], (const float*)d_in[7] };
    const float* Adst[2]= { (const float*)d_in[4], (const float*)d_in[8] };
    const float* B[2]   = { (const float*)d_in[5], (const float*)d_in[9] };

    const size_t NC = (size_t)NNODES * CFEAT;         // 3,200,000 floats
    const size_t NPAD = 50048;                        // padded node count
    const size_t EPAD = 1050112;                      // padded edge count
    float*    w0    = (float*)d_ws;
    float*    h     = w0;                             // [N,C]
    float*    accum = w0 + NC;                        // [N,C]
    float*    bufA  = w0 + 2 * NC;                    // [N,C]
    float*    bufB  = w0 + 3 * NC;                    // [N,C]
    float*    as    = w0 + 4 * NC;                    // [N]
    float*    ad    = as + NPAD;
    float*    sden  = ad + NPAD;
    unsigned* mU    = (unsigned*)(sden + NPAD);
    int*      srcI  = (int*)(mU + NPAD);              // [E]
    int*      dstI  = srcI + EPAD;                    // [E]

    const int TB = 256;
    const int gEdge   = (ETOT + TB - 1) / TB;                 // 1 thread/edge
    const int gEdge32 = (int)(((long long)ETOT * 32 + TB - 1) / TB); // 1 wave/edge
    const int gNode   = (NNODES + TB - 1) / TB;
    const int gElem   = (NNODES * CFEAT + TB - 1) / TB;
    const int gGemm   = ((NNODES / 16) + 7) / 8;              // 8 waves/block

    build_edges<<<gEdge, TB, 0, stream>>>(ei, srcI, dstI);

    const float* xcur = x_in;
    for (int step = 0; step < 20; ++step) {
        int l = step & 1;
        float* xnext = (step == 19) ? (float*)d_out : (l == 0 ? bufA : bufB);

        gemm_h      <<<gGemm,   TB, 0, stream>>>(xcur, W[l], h);
        node_prep   <<<gNode,   TB, 0, stream>>>(h, Asrc[l], Adst[l], as, ad, mU, sden, accum);
        edge_max    <<<gEdge,   TB, 0, stream>>>(srcI, dstI, as, ad, mU);
        edge_scatter<<<gEdge32, TB, 0, stream>>>(srcI, dstI, as, ad, mU, h, sden, accum);
        finalize    <<<gElem,   TB, 0, stream>>>(accum, sden, B[l], xnext);

        xcur = xnext;
    }
}